// MiniLLMIndexer_50946902065256
// MI455X (gfx1250) — compile-verified
//
#include <hip/hip_runtime.h>
#include <hip/hip_bf16.h>

// ---------------------------------------------------------------------------
// MiniLLM indexer for MI455X (gfx1250, wave32).
//   scores.mean over heads == (1/NH) * (Q_256 . K_256^T), so we run:
//     K1: fused Q/K projection GEMM  [4096x1024]x[1024x256] via v_wmma_f32_16x16x4_f32
//     K2: scores GEMM per batch      [2048x256]x[256x2048]  via v_wmma_f32_16x16x4_f32 (+mask pad)
//     K3: per-row top-1024 via LDS bitonic sort (value desc, index asc tiebreak)
// Workspace: Q (4MB) + K (4MB) + scores (33.5MB) -> scores stay L2-resident (192MB L2).
// ---------------------------------------------------------------------------

typedef float v2f __attribute__((ext_vector_type(2)));
typedef float v8f __attribute__((ext_vector_type(8)));

#define BATCH 2
#define SEQ   2048
#define HID   1024
#define DPROJ 256          // NH*HD
#define TOPK  1024
#define ROWS  (BATCH * SEQ)            // 4096
#define SCORE_FACTOR 0.022097086912079608f   // (HD^-0.5) / NH

// ---------------------------------------------------------------------------
// Kernel 1: Q = X*Wq^T, K = X*Wk^T. One wave per 16x16 output tile; A (the X
// tile) is shared between the Q and K accumulators.
// ---------------------------------------------------------------------------
__global__ __launch_bounds__(256)
void qk_proj_kernel(const float* __restrict__ x,
                    const float* __restrict__ wq,
                    const float* __restrict__ wk,
                    float* __restrict__ qout,
                    float* __restrict__ kout) {
    const int lane = threadIdx.x & 31;
    const int warp = threadIdx.x >> 5;
    const int w    = blockIdx.x * 8 + warp;      // 4096 waves total
    const int mt   = w >> 4;                     // 0..255 row tiles
    const int nt   = w & 15;                     // 0..15  col tiles
    const int half = lane >> 4;                  // K half select (ISA A layout)
    const int l    = lane & 15;

    const float* xa = x  + (size_t)(mt * 16 + l) * HID + 2 * half;
    const float* bq = wq + (size_t)(nt * 16 + l) * HID + 2 * half;
    const float* bk = wk + (size_t)(nt * 16 + l) * HID + 2 * half;

    v8f accq = {};
    v8f acck = {};
#pragma unroll 4
    for (int k0 = 0; k0 < HID; k0 += 4) {
        v2f a  = *(const v2f*)(xa + k0);
        v2f qb = *(const v2f*)(bq + k0);
        v2f kb = *(const v2f*)(bk + k0);
        accq = __builtin_amdgcn_wmma_f32_16x16x4_f32(false, a, false, qb,
                                                     (short)0, accq, false, false);
        acck = __builtin_amdgcn_wmma_f32_16x16x4_f32(false, a, false, kb,
                                                     (short)0, acck, false, false);
    }

    const int n = nt * 16 + l;
#pragma unroll
    for (int r = 0; r < 8; ++r) {
        const int m = mt * 16 + half * 8 + r;    // C layout: VGPR r -> M=r / M=r+8
        qout[(size_t)m * DPROJ + n] = accq[r];
        kout[(size_t)m * DPROJ + n] = acck[r];
    }
}

// ---------------------------------------------------------------------------
// Kernel 2: scores[b] = factor * Q[b] * K[b]^T + pad(mask). One wave per
// 16x16 score tile, K-dim = 256.
// ---------------------------------------------------------------------------
__global__ __launch_bounds__(256)
void scores_kernel(const float* __restrict__ q,
                   const float* __restrict__ k,
                   const float* __restrict__ mask,
                   float* __restrict__ scores) {
    const int lane = threadIdx.x & 31;
    const int warp = threadIdx.x >> 5;
    const int w    = blockIdx.x * 8 + warp;      // 32768 waves total
    const int b    = w >> 14;                    // 16384 tiles per batch
    const int rem  = w & 16383;
    const int mt   = rem >> 7;                   // 0..127
    const int nt   = rem & 127;                  // 0..127
    const int half = lane >> 4;
    const int l    = lane & 15;

    const float* qa = q + (size_t)(b * SEQ + mt * 16 + l) * DPROJ + 2 * half;
    const float* kb = k + (size_t)(b * SEQ + nt * 16 + l) * DPROJ + 2 * half;

    v8f acc = {};
#pragma unroll 8
    for (int k0 = 0; k0 < DPROJ; k0 += 4) {
        v2f a  = *(const v2f*)(qa + k0);
        v2f bb = *(const v2f*)(kb + k0);
        acc = __builtin_amdgcn_wmma_f32_16x16x4_f32(false, a, false, bb,
                                                    (short)0, acc, false, false);
    }

    const int   n   = nt * 16 + l;
    const float pad = (1.0f - mask[b * SEQ + n]) * -1e9f;
    float* srow = scores + (size_t)b * SEQ * SEQ;
#pragma unroll
    for (int r = 0; r < 8; ++r) {
        const int m = mt * 16 + half * 8 + r;
        srow[(size_t)m * SEQ + n] = acc[r] * SCORE_FACTOR + pad;
    }
}

// ---------------------------------------------------------------------------
// Kernel 3: top-1024 of each 2048-score row. One 256-thread workgroup per row,
// full bitonic sort in LDS (16KB), descending by value with ascending-index
// tie-break (matches jax.lax.top_k stability). Emit first 1024 indices.
// ---------------------------------------------------------------------------
__global__ __launch_bounds__(256)
void topk_kernel(const float* __restrict__ scores, int* __restrict__ out) {
    __shared__ float sv[SEQ];
    __shared__ int   si[SEQ];

    const int row = blockIdx.x;                       // 0..4095
    const float* srow = scores + (size_t)row * SEQ;

    for (int i = threadIdx.x; i < SEQ; i += 256) {
        sv[i] = srow[i];
        si[i] = i;
    }
    __syncthreads();

    for (int kk = 2; kk <= SEQ; kk <<= 1) {
        for (int j = kk >> 1; j > 0; j >>= 1) {
            for (int i = threadIdx.x; i < SEQ; i += 256) {
                const int ixj = i ^ j;
                if (ixj > i) {
                    const float v0 = sv[i], v1 = sv[ixj];
                    const int   a0 = si[i], a1 = si[ixj];
                    // "before" == v0 precedes v1 in final order (desc value, asc idx)
                    const bool before = (v0 > v1) || (v0 == v1 && a0 < a1);
                    const bool asc    = (i & kk) == 0;
                    if (asc ? !before : before) {
                        sv[i] = v1; sv[ixj] = v0;
                        si[i] = a1; si[ixj] = a0;
                    }
                }
            }
            __syncthreads();
        }
    }

    for (int i = threadIdx.x; i < TOPK; i += 256)
        out[(size_t)row * TOPK + i] = si[i];
}

// ---------------------------------------------------------------------------
extern "C" void kernel_launch(void* const* d_in, const int* in_sizes, int n_in,
                              void* d_out, int out_size, void* d_ws, size_t ws_size,
                              hipStream_t stream) {
    const float* hs   = (const float*)d_in[0];   // [B,S,HID]
    const float* mask = (const float*)d_in[1];   // [B,S]
    const float* wq   = (const float*)d_in[2];   // [DPROJ,HID]
    const float* wk   = (const float*)d_in[3];   // [DPROJ,HID]
    // d_in[4] = past_len (unused, == 0)

    float* wsf     = (float*)d_ws;
    float* q_ws    = wsf;                                   // 4096*256 floats
    float* k_ws    = wsf + (size_t)ROWS * DPROJ;            // 4096*256 floats
    float* sc_ws   = wsf + (size_t)2 * ROWS * DPROJ;        // 2*2048*2048 floats
    int*   out     = (int*)d_out;                           // [B,S,TOPK] int32

    // K1: 4096 tiles (256 mt x 16 nt), 8 waves/block -> 512 blocks
    qk_proj_kernel<<<512, 256, 0, stream>>>(hs, wq, wk, q_ws, k_ws);

    // K2: 32768 tiles (2 b x 128 mt x 128 nt), 8 waves/block -> 4096 blocks
    scores_kernel<<<4096, 256, 0, stream>>>(q_ws, k_ws, mask, sc_ws);

    // K3: one workgroup per score row
    topk_kernel<<<ROWS, 256, 0, stream>>>(sc_ws, out);
}